// RelativePositionalEncoder_1632087573004
// MI455X (gfx1250) — compile-verified
//
#include <hip/hip_runtime.h>

typedef __attribute__((ext_vector_type(2))) float v2f;
typedef __attribute__((ext_vector_type(8))) float v8f;

#define T_Q     1024
#define D       128
#define NREL    257     // 2*128 + 1
#define NTILES  17      // ceil(257/16)
#define NPAD    (NTILES * 16)   // 272
#define QSTRIDE 132     // 128 + 4 pad -> conflict-free A-fragment ds loads

__global__ __launch_bounds__(256)
void relpe_wmma_kernel(const float* __restrict__ Q,
                       const float* __restrict__ pe,
                       float* __restrict__ out)
{
    __shared__ float qs[16 * QSTRIDE];   // 16x128 Q tile (padded)
    __shared__ float rel[16 * NPAD];     // 16x272 rel tile

    const int tid  = threadIdx.x;
    const int lane = tid & 31;
    const int wave = tid >> 5;

    const long long row0 = (long long)blockIdx.x * 16;   // global M row
    const int t0 = (int)(row0 & (T_Q - 1));              // t of first row (16 | 1024)

    // ---- Stage 1: Q tile (16 x 128 f32) -> LDS, 512 float4 loads over 256 threads
    {
        const float* qbase = Q + row0 * D;
        #pragma unroll
        for (int i = 0; i < 2; ++i) {
            const int li = tid + i * 256;       // 0..511 float4 index
            const int r  = li >> 5;             // row 0..15 (32 float4 per row)
            const int c4 = li & 31;             // float4 col
            const float4 v = *(const float4*)(qbase + (size_t)r * D + c4 * 4);
            *(float4*)&qs[r * QSTRIDE + c4 * 4] = v;
        }
    }
    __syncthreads();

    // ---- Stage 2: rel tile = Q_tile (16x128) @ pe^T (128x272) via WMMA f32 16x16x4
    {
        const int j     = lane & 15;        // A: M row / B: N col
        const int khalf = (lane >> 4) * 2;  // 0 or 2 (K sub-offset per ISA layout)

        for (int tile = wave; tile < NTILES; tile += 8) {
            const int n0 = tile * 16;
            const int r  = n0 + j;                          // pe row index
            const float  mask  = (r < NREL) ? 1.0f : 0.0f;  // zero-pad N > 256
            const int    rc    = (r < NREL) ? r : (NREL - 1);
            const float* pe_row = pe + (size_t)rc * D;
            const float* a_row  = &qs[j * QSTRIDE];

            v8f c = {};
            #pragma unroll
            for (int kk = 0; kk < 32; ++kk) {
                const int kbase = kk * 4 + khalf;
                v2f a, b;
                a.x = a_row[kbase];
                a.y = a_row[kbase + 1];
                b.x = pe_row[kbase]     * mask;
                b.y = pe_row[kbase + 1] * mask;
                c = __builtin_amdgcn_wmma_f32_16x16x4_f32(
                        /*neg_a=*/false, a, /*neg_b=*/false, b,
                        /*c_mod=*/(short)0, c, /*reuse_a=*/false, /*reuse_b=*/false);
            }
            // C/D layout: VGPR jv -> lanes 0-15: M=jv, lanes 16-31: M=jv+8; N=lane%16
            const int mbase = (lane < 16) ? 0 : 8;
            #pragma unroll
            for (int jv = 0; jv < 8; ++jv) {
                rel[(mbase + jv) * NPAD + n0 + j] = c[jv];
            }
        }
    }
    __syncthreads();

    // ---- Stage 3: gather + stream 16x1024 outputs (coalesced float4 stores)
    {
        const int lr  = tid >> 4;           // local row 0..15
        const int sub = tid & 15;           // 16 threads per row
        const int t   = t0 + lr;
        float* orow = out + (row0 + lr) * (long long)T_Q;
        const float* relrow = &rel[lr * NPAD];

        #pragma unroll
        for (int i = 0; i < 16; ++i) {
            const int s = i * 64 + sub * 4;
            float4 v;
            #pragma unroll
            for (int k = 0; k < 4; ++k) {
                int dlt = t - (s + k);
                dlt = dlt > 128 ? 128 : (dlt < -128 ? -128 : dlt);
                ((float*)&v)[k] = relrow[dlt + 128];
            }
            *(float4*)&orow[s] = v;
        }
    }
}

extern "C" void kernel_launch(void* const* d_in, const int* in_sizes, int n_in,
                              void* d_out, int out_size, void* d_ws, size_t ws_size,
                              hipStream_t stream)
{
    (void)n_in; (void)out_size; (void)d_ws; (void)ws_size;
    const float* Q  = (const float*)d_in[0];   // (4,16,1024,128) f32
    // d_in[1] = K : only its t-extent (==1024) matters; unused numerically
    const float* pe = (const float*)d_in[2];   // (257,128) f32
    float* out = (float*)d_out;                // (4,16,1024,1024) f32

    const int rows = in_sizes[0] / D;          // 65536
    const int grid = rows / 16;                // 4096 blocks
    relpe_wmma_kernel<<<grid, 256, 0, stream>>>(Q, pe, out);
}